// DDM_64235530879483
// MI455X (gfx1250) — compile-verified
//
#include <hip/hip_runtime.h>
#include <math.h>
#include <stdint.h>

// ---- problem constants -------------------------------------------------
#define NMESH  360          // circular color mesh
#define NPAD   384          // padded to 24 * 16 for WMMA tiling
#define NBATCH 256
#define NBITS  24           // time quantization: t = n * TAU, n < 2^24
#define LDP    256          // leading dim (cols) of P panels
#define TAU    (1000.0f / 16777216.0f)   // T_MAX / 2^24

// block-GEMM tiling
#define MT 64               // block tile M
#define NT 64               // block tile N
#define KC 16               // K chunk staged in LDS
#define NCH (NPAD / KC)     // 24 chunks
#define SROW 72             // padded LDS row stride (words): 2*72 mod 64 = 16 -> halves hit disjoint banks
#define LSZ (KC * SROW)     // one LDS panel (words)

typedef __attribute__((ext_vector_type(2))) float v2f;
typedef __attribute__((ext_vector_type(8))) float v8f;
typedef int v4i __attribute__((vector_size(16)));   // matches builtin's int4 type

// CDNA5 async global->LDS copy path (ASYNCcnt-tracked)
#if defined(__has_builtin)
#if __has_builtin(__builtin_amdgcn_global_load_async_to_lds_b32) && \
    __has_builtin(__builtin_amdgcn_global_load_async_to_lds_b128)
#define HAVE_ASYNC_LDS 1
#endif
#endif

// pointer casts to the builtins' exact parameter types (from hipcc diagnostics)
#define ASG32(p)  ((__attribute__((address_space(1))) int*)(uintptr_t)(p))
#define ASL32(p)  ((__attribute__((address_space(3))) int*)(unsigned)(uintptr_t)(p))
#define ASG128(p) ((__attribute__((address_space(1))) v4i*)(uintptr_t)(p))
#define ASL128(p) ((__attribute__((address_space(3))) v4i*)(unsigned)(uintptr_t)(p))

// 1 / (2*pi*i0e(400)) via the asymptotic series for e^{-z} I0(z)
__device__ __forceinline__ float vm_norm() {
    const double z = 400.0;                       // kappa = 1/SIGMA_INIT^2
    const double PI_D = 3.14159265358979323846;
    const double s = 1.0 + 1.0/(8.0*z) + 9.0/(128.0*z*z)
                   + 225.0/(3072.0*z*z*z) + 11025.0/(98304.0*z*z*z*z);
    const double i0e = s / sqrt(2.0*PI_D*z);
    return (float)(1.0 / (2.0*PI_D*i0e));
}

// ---- T = tau * evo_mat (tridiagonal circulant, zero-padded to 384) -----
__global__ void k_build_T(const float* __restrict__ cm, const float* __restrict__ g,
                          const float* __restrict__ sig, float* __restrict__ T) {
    int t = blockIdx.x * blockDim.x + threadIdx.x;
    if (t >= NPAD * NPAD) return;
    int i = t / NPAD, j = t % NPAD;
    float val = 0.f;
    if (i < NMESH && j < NMESH) {
        float d  = cm[1] - cm[0];
        float s2 = sig[0] * sig[0];
        float coefD = s2 / (2.f * d * d);
        int ip1 = (i + 1) % NMESH, im1 = (i + NMESH - 1) % NMESH;
        if      (j == im1) val = coefD - g[j] / (2.f * d);
        else if (j == ip1) val = coefD + g[j] / (2.f * d);
        else if (j == i)   val = -2.f * coefD;
    }
    T[t] = val * TAU;
}

// ---- X = alpha*T + I ---------------------------------------------------
__global__ void k_axpyI(const float* __restrict__ T, float* __restrict__ X, float alpha) {
    int t = blockIdx.x * blockDim.x + threadIdx.x;
    if (t >= NPAD * NPAD) return;
    int i = t / NPAD, j = t % NPAD;
    X[t] = alpha * T[t] + ((i == j) ? 1.f : 0.f);
}

__device__ __forceinline__ void drain_stage() {
#ifdef HAVE_ASYNC_LDS
    asm volatile("s_wait_asynccnt 0" ::: "memory");  // own copies landed in LDS
#endif
    __syncthreads();                                 // all waves' copies landed
}

// ---- LDS-staged 64x64 block GEMM core ---------------------------------
// 4 waves, 4 accumulators/wave, double-buffered async K-chunk pipeline.
// Buffer select is an INTEGER offset on one __shared__ array so every access
// keeps addrspace(3) provenance and lowers to ds_load / async-to-LDS.
__device__ __forceinline__ void gemm_block(const float* __restrict__ A,
                                           const float* __restrict__ B,
                                           int i0, int j0, int ncols,
                                           v8f acc[4]) {
    __shared__ float ls[4 * LSZ];        // [buf0: A|B][buf1: A|B], 18 KB

    const int tid  = threadIdx.x;
    const int wave = tid >> 5, lane = tid & 31;
    const int hf = lane >> 4, l16 = lane & 15;

    const int am  = tid >> 1;            // A row this thread stages (0..63)
    const int ak0 = (tid & 1) * 8;       // A k-subchunk (0 or 8)
    const int bk  = tid >> 3;            // B k-row this thread stages (0..15)
    const int bc  = (tid & 7) * 8;       // B col start (0..56)

    const float* Abase = A + (size_t)(i0 + am) * NPAD + ak0;
    const float* Bbase = B + (size_t)bk * ncols + j0 + bc;

    const int aoffA = ak0 * SROW + am;   // this thread's A panel base index
    const int boffB = bk * SROW + bc;    // this thread's B panel base index

    // stage chunk c into buffer (c & 1)
    auto stage = [&](int c) {
        const float* srcA = Abase + c * KC;
        const float* srcB = Bbase + (size_t)c * KC * ncols;
        const int off = (c & 1) * 2 * LSZ;
#ifdef HAVE_ASYNC_LDS
        // async element-scatter transpose of A (b32 lanes), bulk copy of B (b128 lanes)
        #pragma unroll
        for (int u = 0; u < 8; ++u)
            __builtin_amdgcn_global_load_async_to_lds_b32(
                ASG32(srcA + u), ASL32(&ls[off + aoffA + u * SROW]), 0, 0);
        __builtin_amdgcn_global_load_async_to_lds_b128(
            ASG128(srcB), ASL128(&ls[off + LSZ + boffB]), 0, 0);
        __builtin_amdgcn_global_load_async_to_lds_b128(
            ASG128(srcB + 4), ASL128(&ls[off + LSZ + boffB + 4]), 0, 0);
#else
        #pragma unroll
        for (int u = 0; u < 8; ++u)
            ls[off + aoffA + u * SROW] = srcA[u];
        #pragma unroll
        for (int u = 0; u < 8; ++u)
            ls[off + LSZ + boffB + u] = srcB[u];
#endif
    };

    stage(0);
    drain_stage();

    for (int c = 0; c < NCH; ++c) {
        const int cur = (c & 1) * 2 * LSZ;
        if (c + 1 < NCH)                 // overlap: DMA next chunk during compute
            stage(c + 1);

        #pragma unroll
        for (int kk = 0; kk < KC; kk += 4) {
            const int ka = kk + hf * 2;
            v2f a;
            a.x = ls[cur + ka * SROW + wave * 16 + l16];
            a.y = ls[cur + (ka + 1) * SROW + wave * 16 + l16];
            #pragma unroll
            for (int n = 0; n < 4; ++n) {
                v2f b;
                b.x = ls[cur + LSZ + ka * SROW + n * 16 + l16];
                b.y = ls[cur + LSZ + (ka + 1) * SROW + n * 16 + l16];
                // v_wmma_f32_16x16x4_f32: D = A(16x4) x B(4x16) + C
                acc[n] = __builtin_amdgcn_wmma_f32_16x16x4_f32(
                             false, a, false, b, (short)0, acc[n], false, false);
            }
        }
        drain_stage();                   // next chunk ready; reads of cur done
    }
}

// ---- generic GEMM: C = alpha*(A@B) [+ I] -------------------------------
__global__ __launch_bounds__(128)
void k_gemm(const float* __restrict__ A, const float* __restrict__ B,
            float* __restrict__ C, int ncols, float alpha, int addI) {
    const int i0 = blockIdx.x * MT, j0 = blockIdx.y * NT;
    const int lane = threadIdx.x & 31, wave = threadIdx.x >> 5;
    const int hf = lane >> 4, l16 = lane & 15;
    v8f acc[4];
    #pragma unroll
    for (int n = 0; n < 4; ++n) acc[n] = (v8f){0.f,0.f,0.f,0.f,0.f,0.f,0.f,0.f};

    gemm_block(A, B, i0, j0, ncols, acc);

    const int rbase = i0 + wave * 16 + hf * 8;   // C/D layout: vgpr r -> row rbase+r
    #pragma unroll
    for (int n = 0; n < 4; ++n) {
        const int col = j0 + n * 16 + l16;
        #pragma unroll
        for (int r = 0; r < 8; ++r) {
            float v = alpha * acc[n][r];
            if (addI && (rbase + r) == col) v += 1.f;
            C[(size_t)(rbase + r) * ncols + col] = v;
        }
    }
}

// ---- per-bit apply: Pout[:,b] = bit_k(n_b) ? M_k @ Pin[:,b] : Pin[:,b] -
__global__ __launch_bounds__(128)
void k_apply(const float* __restrict__ Mk, const float* __restrict__ Pin,
             float* __restrict__ Pout, const unsigned* __restrict__ nArr, int bit) {
    const int i0 = blockIdx.x * MT, j0 = blockIdx.y * NT;
    const int lane = threadIdx.x & 31, wave = threadIdx.x >> 5;
    const int hf = lane >> 4, l16 = lane & 15;
    v8f acc[4];
    #pragma unroll
    for (int n = 0; n < 4; ++n) acc[n] = (v8f){0.f,0.f,0.f,0.f,0.f,0.f,0.f,0.f};

    gemm_block(Mk, Pin, i0, j0, LDP, acc);

    const int rbase = i0 + wave * 16 + hf * 8;
    #pragma unroll
    for (int n = 0; n < 4; ++n) {
        const int col = j0 + n * 16 + l16;
        const bool take = (nArr[col] >> bit) & 1u;
        #pragma unroll
        for (int r = 0; r < 8; ++r) {
            float v = take ? acc[n][r] : Pin[(size_t)(rbase + r) * LDP + col];
            Pout[(size_t)(rbase + r) * LDP + col] = v;
        }
    }
}

// ---- p0[j,b] = vonmises(c_mesh[j]; init_color[b], kappa=400) -----------
__global__ void k_init_p0(const float* __restrict__ cm, const float* __restrict__ ic,
                          float* __restrict__ P) {
    int t = blockIdx.x * blockDim.x + threadIdx.x;
    if (t >= NPAD * NBATCH) return;
    int j = t / NBATCH, b = t % NBATCH;
    float v = 0.f;
    if (j < NMESH) {
        float z = cosf(cm[j] - ic[b]) - 1.f;     // in [-2, 0], stable
        v = expf(400.f * z) * vm_norm();
    }
    P[t] = v;
}

// ---- per-sample metadata: quantized step count + nearest mesh index ----
__global__ void k_meta(const float* __restrict__ cm, const float* __restrict__ dt,
                       const float* __restrict__ rep, unsigned* __restrict__ nArr,
                       int* __restrict__ idxArr) {
    int b = threadIdx.x;
    if (b >= NBATCH) return;
    float nf = dt[b] * (16777216.0f / 1000.0f) + 0.5f;
    long n = (long)nf;
    if (n < 0) n = 0;
    if (n > 16777215) n = 16777215;
    nArr[b] = (unsigned)n;
    float r = rep[b], best = 1e30f; int bi = 0;
    for (int j = 0; j < NMESH; ++j) {            // first strict min == jnp.argmin
        float dd = fabsf(cm[j] - r);
        if (dd < best) { best = dd; bi = j; }
    }
    idxArr[b] = bi;
}

// ---- loss = -mean(log(relu(p_sel) + 1e-5)) -----------------------------
__global__ void k_loss(const float* __restrict__ P, const int* __restrict__ idxArr,
                       float* __restrict__ out) {
    __shared__ float red[NBATCH];
    int b = threadIdx.x;
    float v = P[(size_t)idxArr[b] * LDP + b];
    red[b] = logf(fmaxf(v, 0.f) + 1e-5f);
    __syncthreads();
    for (int s = NBATCH / 2; s > 0; s >>= 1) {
        if (b < s) red[b] += red[b + s];
        __syncthreads();
    }
    if (b == 0) out[0] = -red[0] / (float)NBATCH;
}

extern "C" void kernel_launch(void* const* d_in, const int* in_sizes, int n_in,
                              void* d_out, int out_size, void* d_ws, size_t ws_size,
                              hipStream_t stream) {
    (void)in_sizes; (void)n_in; (void)out_size; (void)ws_size;
    const float* cm  = (const float*)d_in[0];   // c_mesh [360]
    const float* g   = (const float*)d_in[1];   // gtheta [360]
    const float* sig = (const float*)d_in[2];   // sigma_diff [1]
    const float* ic  = (const float*)d_in[3];   // init_color [256]
    const float* dt  = (const float*)d_in[4];   // delay_t [256]
    const float* rep = (const float*)d_in[5];   // report_color [256]
    float* out = (float*)d_out;

    // workspace: T | M[0..23] | Pa | Pb | nArr | idxArr   (~15 MB)
    float* ws = (float*)d_ws;
    const size_t S = (size_t)NPAD * NPAD;
    float* T  = ws;
    float* M  = ws + S;
    float* Pa = M + (size_t)NBITS * S;
    float* Pb = Pa + (size_t)NPAD * NBATCH;
    unsigned* nArr = (unsigned*)(Pb + (size_t)NPAD * NBATCH);
    int* idxArr    = (int*)(nArr + NBATCH);

    const dim3 blk(256);
    const dim3 tb(128);                            // 4 waves per block
    const dim3 gsq(NPAD / MT, NPAD / NT);          // 6x6 blocks, square GEMM
    const dim3 gap(NPAD / MT, NBATCH / NT);        // 6x4 blocks, panel GEMM

    // T = tau * evo_mat
    k_build_T<<<(NPAD * NPAD + 255) / 256, blk, 0, stream>>>(cm, g, sig, T);

    // B = exp(T) via order-4 Horner: I + T(I + T/2(I + T/3(I + T/4)))
    k_axpyI<<<(NPAD * NPAD + 255) / 256, blk, 0, stream>>>(T, M + S, 0.25f);
    k_gemm<<<gsq, tb, 0, stream>>>(T, M + S,     M + 2 * S, NPAD, 1.f / 3.f, 1);
    k_gemm<<<gsq, tb, 0, stream>>>(T, M + 2 * S, M + S,     NPAD, 0.5f,      1);
    k_gemm<<<gsq, tb, 0, stream>>>(T, M + S,     M,         NPAD, 1.f,       1);

    // M_k = B^(2^k) : 23 shared squarings (whole batch reuses these)
    for (int k = 1; k < NBITS; ++k)
        k_gemm<<<gsq, tb, 0, stream>>>(M + (size_t)(k - 1) * S,
                                       M + (size_t)(k - 1) * S,
                                       M + (size_t)k * S, NPAD, 1.f, 0);

    // p0 panel (384x256, padded rows zero) + per-sample metadata
    k_init_p0<<<(NPAD * NBATCH + 255) / 256, blk, 0, stream>>>(cm, ic, Pa);
    k_meta<<<1, NBATCH, 0, stream>>>(cm, dt, rep, nArr, idxArr);

    // apply exp(t_b A) = prod over set bits of n_b (powers commute)
    float* pin = Pa; float* pout = Pb;
    for (int bit = 0; bit < NBITS; ++bit) {
        k_apply<<<gap, tb, 0, stream>>>(M + (size_t)bit * S, pin, pout, nArr, bit);
        float* tswap = pin; pin = pout; pout = tswap;
    }
    // NBITS even -> final panel back in Pa (== pin)
    k_loss<<<1, NBATCH, 0, stream>>>(pin, idxArr, out);
}